// GCNRecommender_22110491640565
// MI455X (gfx1250) — compile-verified
//
#include <hip/hip_runtime.h>
#include <hip/hip_bf16.h>
#include <math.h>

typedef __attribute__((ext_vector_type(2))) float v2f;
typedef __attribute__((ext_vector_type(8))) float v8f;

static inline int ceil_div_ll(long long a, long long b) { return (int)((a + b - 1) / b); }

// ---------------------------------------------------------------------------
// fp32 WMMA GEMM: C[N,M] = A[N,K] @ W[K,M] (+ optional bias row)
// One wave32 computes one 16x16 output tile via V_WMMA_F32_16X16X4_F32.
// Lane layout per CDNA5 ISA 7.12.2:
//   A 16x4 : lanes 0-15 -> M=lane, VGPR0/1 = K=0/1 ; lanes 16-31 -> K=2/3
//   B 4x16 : VGPR0: lanes0-15 K=0 / lanes16-31 K=2 ; VGPR1: K=1 / K=3
//   C 16x16: VGPR r: lanes0-15 M=r, lanes16-31 M=r+8 ; N=lane%16
// ---------------------------------------------------------------------------
template<int K, int M>
__global__ __launch_bounds__(256)
void wmma_gemm_f32(const float* __restrict__ A, const float* __restrict__ W,
                   const float* __restrict__ bias, float* __restrict__ C, int N)
{
    constexpr int TILES_N = M / 16;
    int wave = (int)(blockIdx.x * (blockDim.x >> 5)) + (int)(threadIdx.x >> 5);
    int lane = threadIdx.x & 31;
    int tile_m = wave / TILES_N;
    int row0 = tile_m * 16;
    if (row0 >= N) return;                 // wave-uniform: EXEC stays all-1s for WMMA
    int col0 = (wave % TILES_N) * 16;
    int lh = lane >> 4;                    // 0: K pair {k,k+1}, 1: {k+2,k+3}
    int lm = lane & 15;

    const float* a_ptr = A + (size_t)(row0 + lm) * K + (lh << 1);
    const float* b_ptr = W + (size_t)(lh << 1) * M + col0 + lm;

    v8f acc = {};
#pragma unroll
    for (int k = 0; k < K; k += 4) {
        v2f a, b;
        a.x = a_ptr[k];
        a.y = a_ptr[k + 1];
        b.x = b_ptr[(size_t)k * M];
        b.y = b_ptr[(size_t)(k + 1) * M];
        acc = __builtin_amdgcn_wmma_f32_16x16x4_f32(false, a, false, b,
                                                    (short)0, acc, false, false);
    }

    float badd = bias ? bias[col0 + lm] : 0.0f;
#pragma unroll
    for (int r = 0; r < 8; ++r) {
        int row = row0 + r + (lh << 3);
        C[(size_t)row * M + col0 + lm] = acc[r] + badd;
    }
}

// ---------------------------------------------------------------------------
__global__ void fill_f32_kernel(float* __restrict__ p, float v, long long n)
{
    long long i = (long long)blockIdx.x * blockDim.x + threadIdx.x;
    if (i < n) p[i] = v;
}

__global__ void deg_kernel(const int* __restrict__ dst, float* __restrict__ deg, int E)
{
    int e = blockIdx.x * blockDim.x + threadIdx.x;
    if (e < E) atomicAdd(&deg[dst[e]], 1.0f);
}

__global__ void dinv_kernel(const float* __restrict__ deg, float* __restrict__ dinv, int n)
{
    int i = blockIdx.x * blockDim.x + threadIdx.x;
    if (i < n) dinv[i] = rsqrtf(deg[i] + 1.0f);   // +1 self loop
}

__global__ void norm_kernel(const int* __restrict__ src, const int* __restrict__ dst,
                            const float* __restrict__ dinv, float* __restrict__ nrm, int E)
{
    int e = blockIdx.x * blockDim.x + threadIdx.x;
    if (e < E) nrm[e] = dinv[src[e]] * dinv[dst[e]];
}

// ---------------------------------------------------------------------------
// Edge scatter: agg[dst] += xw[src] * norm.  Consecutive lanes cover
// consecutive float4 feature chunks of the same edge (coalesced row gather),
// scatter via L2-resident global_atomic_add_f32.
// ---------------------------------------------------------------------------
template<int M>
__global__ __launch_bounds__(256)
void edge_scatter(const int* __restrict__ src, const int* __restrict__ dst,
                  const float* __restrict__ nrm, const float* __restrict__ xw,
                  float* __restrict__ agg, int E)
{
    constexpr int CPE = M / 4;             // float4 chunks per edge (power of two)
    int gid = blockIdx.x * blockDim.x + threadIdx.x;
    int e = gid / CPE;
    if (e >= E) return;
    int c = (gid % CPE) << 2;
    int s = src[e], d = dst[e];
    float nm = nrm[e];
    const float4 v = *(const float4*)(xw + (size_t)s * M + c);
    float* o = agg + (size_t)d * M + c;
    atomicAdd(o + 0, v.x * nm);
    atomicAdd(o + 1, v.y * nm);
    atomicAdd(o + 2, v.z * nm);
    atomicAdd(o + 3, v.w * nm);
}

// ---------------------------------------------------------------------------
// Finalize: h = ELU(LN(agg + xw*dinv^2 + b)). One wave32 per node, shuffle
// reductions across 32 lanes; each lane owns M/32 features.
// ---------------------------------------------------------------------------
template<int M>
__global__ __launch_bounds__(256)
void finalize_ln_elu(const float* __restrict__ agg, const float* __restrict__ xw,
                     const float* __restrict__ dinv, const float* __restrict__ bias,
                     const float* __restrict__ g, const float* __restrict__ be,
                     float* __restrict__ out, int N)
{
    constexpr int PER = M / 32;
    int node = (int)(blockIdx.x * (blockDim.x >> 5)) + (int)(threadIdx.x >> 5);
    int lane = threadIdx.x & 31;
    if (node >= N) return;

    float vals[PER];
    float sum = 0.0f;
    float selfw = xw ? dinv[node] * dinv[node] : 0.0f;
#pragma unroll
    for (int i = 0; i < PER; ++i) {
        int c = lane + (i << 5);
        float v = agg[(size_t)node * M + c];
        if (xw)   v += xw[(size_t)node * M + c] * selfw;
        if (bias) v += bias[c];
        vals[i] = v;
        sum += v;
    }
#pragma unroll
    for (int off = 16; off; off >>= 1) sum += __shfl_xor(sum, off, 32);
    float mu = sum * (1.0f / M);
    float vs = 0.0f;
#pragma unroll
    for (int i = 0; i < PER; ++i) { float d = vals[i] - mu; vs += d * d; }
#pragma unroll
    for (int off = 16; off; off >>= 1) vs += __shfl_xor(vs, off, 32);
    float rstd = rsqrtf(vs * (1.0f / M) + 1e-5f);
#pragma unroll
    for (int i = 0; i < PER; ++i) {
        int c = lane + (i << 5);
        float v = (vals[i] - mu) * rstd * g[c] + be[c];
        out[(size_t)node * M + c] = (v > 0.0f) ? v : (expf(v) - 1.0f);
    }
}

// ---------------------------------------------------------------------------
extern "C" void kernel_launch(void* const* d_in, const int* in_sizes, int n_in,
                              void* d_out, int out_size, void* d_ws, size_t ws_size,
                              hipStream_t stream)
{
    const float* x   = (const float*)d_in[0];
    const int*   ei  = (const int*)  d_in[1];
    const float* W1  = (const float*)d_in[2];
    const float* b1  = (const float*)d_in[3];
    const float* g1  = (const float*)d_in[4];
    const float* be1 = (const float*)d_in[5];
    const float* W2  = (const float*)d_in[6];
    const float* b2  = (const float*)d_in[7];
    const float* g2  = (const float*)d_in[8];
    const float* be2 = (const float*)d_in[9];
    const float* W3  = (const float*)d_in[10];
    const float* b3  = (const float*)d_in[11];
    const float* g3  = (const float*)d_in[12];
    const float* be3 = (const float*)d_in[13];
    const float* lw1 = (const float*)d_in[14];
    const float* lb1 = (const float*)d_in[15];
    const float* g4  = (const float*)d_in[16];
    const float* be4 = (const float*)d_in[17];
    const float* lw2 = (const float*)d_in[18];
    const float* lb2 = (const float*)d_in[19];

    const int N = in_sizes[0] / 128;
    const int E = in_sizes[1] / 2;
    const int* src = ei;
    const int* dst = ei + E;

    // workspace carve-out (~161 MB)
    float* ws   = (float*)d_ws;
    float* xw   = ws;  ws += (size_t)N * 128;
    float* agg  = ws;  ws += (size_t)N * 128;
    float* h    = ws;  ws += (size_t)N * 128;
    float* deg  = ws;  ws += N;
    float* dinv = ws;  ws += N;
    float* nrm  = ws;  ws += E;

    const int B = 256;
    auto gemm_blocks = [N](int tiles_n) {
        long long waves = (long long)((N + 15) / 16) * tiles_n;
        return (int)((waves + 7) / 8);          // 8 waves per 256-thread block
    };
    const int node_blocks = ceil_div_ll(N, 8);  // one wave per node

    // ---- graph-invariant prep: deg -> dinv -> norm[e] ----
    fill_f32_kernel<<<ceil_div_ll(N, B), B, 0, stream>>>(deg, 0.0f, (long long)N);
    deg_kernel<<<ceil_div_ll(E, B), B, 0, stream>>>(dst, deg, E);
    dinv_kernel<<<ceil_div_ll(N, B), B, 0, stream>>>(deg, dinv, N);
    norm_kernel<<<ceil_div_ll(E, B), B, 0, stream>>>(src, dst, dinv, nrm, E);

    // ---- GCN layer 1: 128 -> 64 ----
    wmma_gemm_f32<128, 64><<<gemm_blocks(4), B, 0, stream>>>(x, W1, nullptr, xw, N);
    fill_f32_kernel<<<ceil_div_ll((long long)N * 64, B), B, 0, stream>>>(agg, 0.0f, (long long)N * 64);
    edge_scatter<64><<<ceil_div_ll((long long)E * 16, B), B, 0, stream>>>(src, dst, nrm, xw, agg, E);
    finalize_ln_elu<64><<<node_blocks, B, 0, stream>>>(agg, xw, dinv, b1, g1, be1, h, N);

    // ---- GCN layer 2: 64 -> 128 ----
    wmma_gemm_f32<64, 128><<<gemm_blocks(8), B, 0, stream>>>(h, W2, nullptr, xw, N);
    fill_f32_kernel<<<ceil_div_ll((long long)N * 128, B), B, 0, stream>>>(agg, 0.0f, (long long)N * 128);
    edge_scatter<128><<<ceil_div_ll((long long)E * 32, B), B, 0, stream>>>(src, dst, nrm, xw, agg, E);
    finalize_ln_elu<128><<<node_blocks, B, 0, stream>>>(agg, xw, dinv, b2, g2, be2, h, N);

    // ---- GCN layer 3: 128 -> 64 ----
    wmma_gemm_f32<128, 64><<<gemm_blocks(4), B, 0, stream>>>(h, W3, nullptr, xw, N);
    fill_f32_kernel<<<ceil_div_ll((long long)N * 64, B), B, 0, stream>>>(agg, 0.0f, (long long)N * 64);
    edge_scatter<64><<<ceil_div_ll((long long)E * 16, B), B, 0, stream>>>(src, dst, nrm, xw, agg, E);
    finalize_ln_elu<64><<<node_blocks, B, 0, stream>>>(agg, xw, dinv, b3, g3, be3, h, N);

    // ---- head: linear 64->32 (+bias), LN+ELU, linear 32->32 (+bias) ----
    wmma_gemm_f32<64, 32><<<gemm_blocks(2), B, 0, stream>>>(h, lw1, lb1, xw, N);
    finalize_ln_elu<32><<<node_blocks, B, 0, stream>>>(xw, nullptr, nullptr, nullptr, g4, be4, h, N);
    wmma_gemm_f32<32, 32><<<gemm_blocks(2), B, 0, stream>>>(h, lw2, lb2, (float*)d_out, N);
}